// Decoder_11785390260915
// MI455X (gfx1250) — compile-verified
//
#include <hip/hip_runtime.h>
#include <math.h>

typedef __attribute__((ext_vector_type(2))) float v2f;
typedef __attribute__((ext_vector_type(8))) float v8f;

#define HIDDEN 64
#define GATES  256
#define IN_DIM 14

__device__ __forceinline__ float sigmoidf_(float x) {
    return 1.0f / (1.0f + __expf(-x));
}
__device__ __forceinline__ int imin_(int a, int b) { return a < b ? a : b; }
__device__ __forceinline__ int imax_(int a, int b) { return a > b ? a : b; }

// G0[t][g] = (inputs @ W_ih0^T)[t][g] + b_ih0[g] + b_hh0[g]
// inputs[t] = t==0 ? 0 : [target[t-1] (8) | difficulty (6) | 0 0]  (padded to K=16)
// One wave per 16x16 output tile; K=16 via 4 chained fp32 WMMAs (exact fp32).
// All guard conditions are handled with clamped unconditional loads + selects
// so no exec-mask divergence surrounds the WMMA fragments.
__global__ void __launch_bounds__(256)
g0_precompute_wmma(const float* __restrict__ target,
                   const float* __restrict__ difficulty,
                   const float* __restrict__ W_ih0,   // [256,14]
                   const float* __restrict__ b_ih0,
                   const float* __restrict__ b_hh0,
                   float* __restrict__ G0, int T, int mTiles) {
    int wave = (int)((blockIdx.x * blockDim.x + threadIdx.x) >> 5);
    int lane = (int)(threadIdx.x & 31);
    int mt = wave >> 4;          // 16 N-tiles (256 gates) per M-tile
    int nt = wave & 15;
    if (mt >= mTiles) return;

    int m15   = lane & 15;
    int row   = mt * 16 + m15;          // A-matrix M index (same for both lane halves)
    int n     = nt * 16 + m15;          // B/D N index
    int khalf = (lane >> 4) * 2;        // lanes 16-31 hold K+2, K+3

    const int  rowc  = imin_(imax_(row, 1), T - 1) - 1;  // clamped target row (row-1)
    const bool rowOK = (row >= 1) && (row < T);

    v8f acc = {};
    #pragma unroll
    for (int kb = 0; kb < 16; kb += 4) {
        int k0 = kb + khalf;
        int k1 = k0 + 1;
        // --- A fragment (teacher-forced input), branchless ---
        float t0 = target[rowc * 8 + (k0 & 7)];
        float t1 = target[rowc * 8 + (k1 & 7)];
        float d0 = difficulty[imin_(imax_(k0 - 8, 0), 5)];
        float d1 = difficulty[imin_(imax_(k1 - 8, 0), 5)];
        float a0 = (k0 < 8) ? t0 : ((k0 < IN_DIM) ? d0 : 0.0f);
        float a1 = (k1 < 8) ? t1 : ((k1 < IN_DIM) ? d1 : 0.0f);
        a0 = rowOK ? a0 : 0.0f;
        a1 = rowOK ? a1 : 0.0f;
        // --- B fragment (W_ih0^T column), branchless ---
        float w0 = W_ih0[n * IN_DIM + imin_(k0, IN_DIM - 1)];
        float w1 = W_ih0[n * IN_DIM + imin_(k1, IN_DIM - 1)];
        v2f a, b;
        a.x = a0;                         a.y = a1;
        b.x = (k0 < IN_DIM) ? w0 : 0.0f;  b.y = (k1 < IN_DIM) ? w1 : 0.0f;
        acc = __builtin_amdgcn_wmma_f32_16x16x4_f32(false, a, false, b,
                                                    (short)0, acc, false, false);
    }
    float bias  = b_ih0[n] + b_hh0[n];
    int   mbase = mt * 16 + (lane >> 4) * 8;    // C/D layout: VGPR j -> M = j (+8 hi lanes)
    #pragma unroll
    for (int j = 0; j < 8; ++j) {
        int r = mbase + j;
        if (r < T) G0[(size_t)r * GATES + n] = acc[j] + bias;
    }
}

// Persistent single-workgroup LSTM recurrence. 256 threads = 8 wave32 waves.
// Thread t owns gate row t of both layers. Cell states c0/c1 live in registers.
// 4 workgroup barriers per step; the FC heads (lanes 0-31 -> lanes 0-7) are
// wave-0-local, so they need only a compiler scheduling fence (same-wave LDS
// ops are in-order per the DScnt rules).
__global__ void __launch_bounds__(256)
decoder_seq(const float* __restrict__ G0,        // [T,256] or nullptr
            const float* __restrict__ target,
            const float* __restrict__ difficulty,
            const float* __restrict__ W_ih0, const float* __restrict__ b_ih0,
            const float* __restrict__ b_hh0,
            const float* __restrict__ h0_in, const float* __restrict__ c0_in,
            const float* __restrict__ W_hh0,
            const float* __restrict__ W_ih1, const float* __restrict__ W_hh1,
            const float* __restrict__ b_ih1, const float* __restrict__ b_hh1,
            const float* __restrict__ W_hid, const float* __restrict__ b_hid,
            const float* __restrict__ W_out, const float* __restrict__ b_out,
            float* __restrict__ out, int T) {
    __shared__ float sW1[GATES * 128];   // [W_ih1 | W_hh1] rows, 128 KB
    __shared__ float sWhid[32 * 64];     // 8 KB
    __shared__ float sWout[8 * 32];      // 1 KB
    __shared__ float sGate[GATES];
    __shared__ float sH0[HIDDEN];
    __shared__ float sH1[HIDDEN];
    __shared__ float sHid[32];
    __shared__ float sDiff[6];

    const int tid = (int)threadIdx.x;

    // ---- stage weights on-chip (once) ----
    for (int i = tid; i < GATES * 128; i += 256) {
        int r = i >> 7, c = i & 127;
        sW1[i] = (c < 64) ? W_ih1[r * 64 + c] : W_hh1[r * 64 + (c - 64)];
    }
    for (int i = tid; i < 32 * 64; i += 256) sWhid[i] = W_hid[i];
    for (int i = tid; i < 8 * 32;  i += 256) sWout[i] = W_out[i];
    if (tid < HIDDEN) { sH0[tid] = h0_in[tid]; sH1[tid] = h0_in[HIDDEN + tid]; }
    if (tid < 6) sDiff[tid] = difficulty[tid];

    float wh0r[HIDDEN];                  // W_hh0 row in registers
    #pragma unroll
    for (int k = 0; k < HIDDEN; ++k) wh0r[k] = W_hh0[tid * HIDDEN + k];
    float wih0r[IN_DIM];                 // W_ih0 row (fallback path)
    #pragma unroll
    for (int k = 0; k < IN_DIM; ++k) wih0r[k] = W_ih0[tid * IN_DIM + k];

    const float b1r    = b_ih1[tid] + b_hh1[tid];
    const float g0bias = b_ih0[tid] + b_hh0[tid];
    float c0r = (tid < HIDDEN) ? c0_in[tid] : 0.0f;
    float c1r = (tid < HIDDEN) ? c0_in[HIDDEN + tid] : 0.0f;
    const float bh = (tid < 32) ? b_hid[tid] : 0.0f;
    const float bo = (tid < 8)  ? b_out[tid] : 0.0f;

    __syncthreads();

    for (int t = 0; t < T; ++t) {
        // ---- layer 0 gates ----
        float g0;
        if (G0) {
            g0 = G0[(size_t)t * GATES + tid];
            if (t + 1 < T)
                __builtin_prefetch(&G0[(size_t)(t + 1) * GATES + tid], 0, 0);
        } else {
            g0 = g0bias;
            if (t > 0) {
                const float* xt = &target[(t - 1) * 8];
                #pragma unroll
                for (int k = 0; k < 8; ++k) g0 += wih0r[k] * xt[k];
                #pragma unroll
                for (int k = 0; k < 6; ++k) g0 += wih0r[8 + k] * sDiff[k];
            }
        }
        #pragma unroll
        for (int k = 0; k < HIDDEN; ++k) g0 += wh0r[k] * sH0[k];
        sGate[tid] = g0;
        __syncthreads();                              // (1)

        if (tid < HIDDEN) {              // LSTM cell 0 (torch order i,f,g,o)
            float gi = sigmoidf_(sGate[tid]);
            float gf = sigmoidf_(sGate[HIDDEN + tid]);
            float gg = tanhf(sGate[2 * HIDDEN + tid]);
            float go = sigmoidf_(sGate[3 * HIDDEN + tid]);
            c0r = gf * c0r + gi * gg;
            sH0[tid] = go * tanhf(c0r);
        }
        __syncthreads();                              // (2)

        // ---- layer 1 gates ----
        float g1 = b1r;
        const float* wr = &sW1[tid * 128];
        #pragma unroll
        for (int k = 0; k < HIDDEN; ++k) g1 += wr[k] * sH0[k];
        #pragma unroll
        for (int k = 0; k < HIDDEN; ++k) g1 += wr[64 + k] * sH1[k];
        sGate[tid] = g1;
        __syncthreads();                              // (3)

        if (tid < HIDDEN) {              // LSTM cell 1
            float gi = sigmoidf_(sGate[tid]);
            float gf = sigmoidf_(sGate[HIDDEN + tid]);
            float gg = tanhf(sGate[2 * HIDDEN + tid]);
            float go = sigmoidf_(sGate[3 * HIDDEN + tid]);
            c1r = gf * c1r + gi * gg;
            sH1[tid] = go * tanhf(c1r);
        }
        __syncthreads();                              // (4)

        // ---- FC heads: entirely within wave 0 (lanes 0-31 -> lanes 0-7) ----
        if (tid < 32) {                  // hiddenfc 64 -> 32
            float a = bh;
            #pragma unroll
            for (int k = 0; k < HIDDEN; ++k) a += sWhid[tid * HIDDEN + k] * sH1[k];
            sHid[tid] = a;
        }
        __builtin_amdgcn_wave_barrier(); // scheduling fence; same-wave LDS is in-order
        if (tid < 8) {                   // outputfc 32 -> 8, round-to-nearest-even
            float a = bo;
            #pragma unroll
            for (int k = 0; k < 32; ++k) a += sWout[tid * 32 + k] * sHid[k];
            out[(size_t)t * 8 + tid] = rintf(a);
        }
        // No barrier: sHid/sGate next writers are ordered by barriers (4)/(1).
    }

    if (tid < HIDDEN) {                  // h_final [2,64] then c_final [2,64]
        size_t base = (size_t)T * 8;
        out[base + tid]              = sH0[tid];
        out[base + HIDDEN + tid]     = sH1[tid];
        out[base + 2 * HIDDEN + tid] = c0r;
        out[base + 3 * HIDDEN + tid] = c1r;
    }
}

extern "C" void kernel_launch(void* const* d_in, const int* in_sizes, int n_in,
                              void* d_out, int out_size, void* d_ws, size_t ws_size,
                              hipStream_t stream) {
    // setup_inputs() order
    const float* h0     = (const float*)d_in[1];
    const float* c0     = (const float*)d_in[2];
    const float* diff   = (const float*)d_in[3];
    const float* target = (const float*)d_in[4];
    const float* W_ih0  = (const float*)d_in[5];
    const float* W_hh0  = (const float*)d_in[6];
    const float* b_ih0  = (const float*)d_in[7];
    const float* b_hh0  = (const float*)d_in[8];
    const float* W_ih1  = (const float*)d_in[9];
    const float* W_hh1  = (const float*)d_in[10];
    const float* b_ih1  = (const float*)d_in[11];
    const float* b_hh1  = (const float*)d_in[12];
    const float* W_hid  = (const float*)d_in[13];
    const float* b_hid  = (const float*)d_in[14];
    const float* W_out  = (const float*)d_in[15];
    const float* b_out  = (const float*)d_in[16];

    const int T = in_sizes[4] / 8;

    size_t need = (size_t)T * GATES * sizeof(float);
    float* G0 = (ws_size >= need) ? (float*)d_ws : nullptr;

    if (G0) {
        int mTiles = (T + 15) / 16;
        int waves  = mTiles * 16;        // one wave per 16x16 tile of G0
        int blocks = (waves + 7) / 8;    // 8 wave32 waves per 256-thread block
        g0_precompute_wmma<<<blocks, 256, 0, stream>>>(target, diff, W_ih0,
                                                       b_ih0, b_hh0, G0, T, mTiles);
    }
    decoder_seq<<<1, 256, 0, stream>>>(G0, target, diff, W_ih0, b_ih0, b_hh0,
                                       h0, c0, W_hh0, W_ih1, W_hh1, b_ih1, b_hh1,
                                       W_hid, b_hid, W_out, b_out,
                                       (float*)d_out, T);
}